// SpatialLineGraph_48902497632460
// MI455X (gfx1250) — compile-verified
//
#include <hip/hip_runtime.h>
#include <hip/hip_bf16.h>
#include <math.h>

typedef __attribute__((ext_vector_type(2))) float v2f;
typedef __attribute__((ext_vector_type(8))) float v8f;

#define WPB 8           // waves per block (256 threads / wave32)
#define MAXD 16         // WMMA tile edge
#define PI_F 3.14159265358979323846f

#if __has_builtin(__builtin_amdgcn_wmma_f32_16x16x4_f32)
#define HAS_WMMA_F32X4 1
#else
#define HAS_WMMA_F32X4 0
#endif

// ---------------- helper kernels ----------------

__global__ void slg_zero_kernel(int* p, int n) {
    int i = blockIdx.x * blockDim.x + threadIdx.x;
    if (i < n) p[i] = 0;
}

__global__ void slg_count_kernel(const int* __restrict__ el, int E,
                                 int* __restrict__ deg_in, int* __restrict__ deg_out) {
    int e = blockIdx.x * blockDim.x + threadIdx.x;
    if (e < E) {
        atomicAdd(&deg_in[el[2 * e + 0]], 1);   // bincount(node_in)
        atomicAdd(&deg_out[el[2 * e + 1]], 1);  // bincount(node_out)
    }
}

// Single-block 3-way exclusive scan: bo_in = exscan(deg_in), bo_out = exscan(deg_out),
// starts = exscan(deg_in*deg_out)
__global__ void slg_scan_kernel(const int* __restrict__ deg_in, const int* __restrict__ deg_out,
                                int N, int* __restrict__ bo_in, int* __restrict__ bo_out,
                                int* __restrict__ starts) {
    __shared__ int part[256][3];
    int t = threadIdx.x;
    int chunk = (N + 255) / 256;
    int lo = t * chunk;
    int hi = min(N, lo + chunk);
    int s0 = 0, s1 = 0, s2 = 0;
    for (int i = lo; i < hi; ++i) {
        int a = deg_in[i], b = deg_out[i];
        s0 += a; s1 += b; s2 += a * b;
    }
    part[t][0] = s0; part[t][1] = s1; part[t][2] = s2;
    __syncthreads();
    if (t == 0) {
        int c0 = 0, c1 = 0, c2 = 0;
        for (int i = 0; i < 256; ++i) {
            int a = part[i][0], b = part[i][1], c = part[i][2];
            part[i][0] = c0; part[i][1] = c1; part[i][2] = c2;
            c0 += a; c1 += b; c2 += c;
        }
    }
    __syncthreads();
    int c0 = part[t][0], c1 = part[t][1], c2 = part[t][2];
    for (int i = lo; i < hi; ++i) {
        int a = deg_in[i], b = deg_out[i];
        bo_in[i] = c0; bo_out[i] = c1; starts[i] = c2;
        c0 += a; c1 += b; c2 += a * b;
    }
}

__global__ void slg_scatter_kernel(const int* __restrict__ el, int E,
                                   const int* __restrict__ bo_in, const int* __restrict__ bo_out,
                                   int* __restrict__ cnt_in, int* __restrict__ cnt_out,
                                   int* __restrict__ sorted_out, int* __restrict__ sorted_in) {
    int e = blockIdx.x * blockDim.x + threadIdx.x;
    if (e < E) {
        int vi = el[2 * e + 0];
        int vo = el[2 * e + 1];
        int p = bo_in[vi] + atomicAdd(&cnt_in[vi], 1);
        sorted_out[p] = e;               // edge_out_sorted = argsort(node_in)
        int q = bo_out[vo] + atomicAdd(&cnt_out[vo], 1);
        sorted_in[q] = e;                // edge_in_sorted = argsort(node_out)
    }
}

__device__ __forceinline__ void slg_insertion_sort(int* a, int n) {
    for (int i = 1; i < n; ++i) {
        int key = a[i];
        int j = i - 1;
        while (j >= 0 && a[j] > key) { a[j + 1] = a[j]; --j; }
        a[j + 1] = key;
    }
}

// Restore stable (= ascending edge id) order inside each bucket.
__global__ void slg_bucketsort_kernel(int N, const int* __restrict__ deg_in,
                                      const int* __restrict__ deg_out,
                                      const int* __restrict__ bo_in, const int* __restrict__ bo_out,
                                      int* __restrict__ sorted_out, int* __restrict__ sorted_in) {
    int v = blockIdx.x * blockDim.x + threadIdx.x;
    if (v < N) {
        slg_insertion_sort(sorted_out + bo_in[v], deg_in[v]);
        slg_insertion_sort(sorted_in + bo_out[v], deg_out[v]);
    }
}

// ---------------- main line-graph kernel: 1 wave == 1 center node ----------------

__global__ void slg_line_kernel(const int* __restrict__ el, const float* __restrict__ pos, int N,
                                const int* __restrict__ deg_in, const int* __restrict__ deg_out,
                                const int* __restrict__ bo_in, const int* __restrict__ bo_out,
                                const int* __restrict__ starts,
                                const int* __restrict__ sorted_out, const int* __restrict__ sorted_in,
                                int* __restrict__ out_edges, float* __restrict__ out_angle,
                                float* __restrict__ out_w) {
    __shared__ float sV1[WPB][MAXD][3];   // outgoing-edge vectors (cols b)
    __shared__ float sV2[WPB][MAXD][3];   // incoming-edge vectors (rows a)
    __shared__ float sN1[WPB][MAXD];      // |v1|^2
    __shared__ float sN2[WPB][MAXD];      // |v2|^2
    __shared__ int   sEout[WPB][MAXD];    // edge ids (edge_out bucket)
    __shared__ int   sEin[WPB][MAXD];     // edge ids (edge_in bucket)

    const int w    = threadIdx.x >> 5;
    const int lane = threadIdx.x & 31;
    const int v    = blockIdx.x * WPB + w;
    const bool valid = v < N;

    int din = 0, dout = 0, boi = 0, boo = 0, st = 0;
    float pjx = 0.f, pjy = 0.f, pjz = 0.f;
    if (valid) {
        din  = deg_in[v];
        dout = deg_out[v];
        boi  = bo_in[v];
        boo  = bo_out[v];
        st   = starts[v];
        pjx = pos[3 * v + 0]; pjy = pos[3 * v + 1]; pjz = pos[3 * v + 2];
    }

    const int  m  = lane & 15;
    const bool hi = lane >= 16;

    // Gather phase: low half-wave loads the din outgoing edges (v1), high half-wave
    // loads the dout incoming edges (v2). Rows/cols beyond the degree are zero-padded.
    if (valid) {
        if (!hi) {
            if (m < din && m < MAXD) {
                int e  = sorted_out[boi + m];
                int ni = el[2 * e + 1];               // node_out[edge_out]
                float x = pos[3 * ni + 0] - pjx;
                float y = pos[3 * ni + 1] - pjy;
                float z = pos[3 * ni + 2] - pjz;
                sV1[w][m][0] = x; sV1[w][m][1] = y; sV1[w][m][2] = z;
                sN1[w][m] = x * x + y * y + z * z;
                sEout[w][m] = e;
            } else if (m < MAXD) {
                sV1[w][m][0] = 0.f; sV1[w][m][1] = 0.f; sV1[w][m][2] = 0.f;
                sN1[w][m] = 0.f; sEout[w][m] = 0;
            }
        } else {
            if (m < dout && m < MAXD) {
                int e  = sorted_in[boo + m];
                int nk = el[2 * e + 0];               // node_in[edge_in]
                float x = pos[3 * nk + 0] - pjx;
                float y = pos[3 * nk + 1] - pjy;
                float z = pos[3 * nk + 2] - pjz;
                sV2[w][m][0] = x; sV2[w][m][1] = y; sV2[w][m][2] = z;
                sN2[w][m] = x * x + y * y + z * z;
                sEin[w][m] = e;
            } else if (m < MAXD) {
                sV2[w][m][0] = 0.f; sV2[w][m][1] = 0.f; sV2[w][m][2] = 0.f;
                sN2[w][m] = 0.f; sEin[w][m] = 0;
            }
        }
    }
    __syncthreads();

    const bool useWmma = valid && din > 0 && dout > 0 && din <= MAXD && dout <= MAXD;

    if (useWmma) {
        // x[a][b] = dot(v2_a, v1_b): one V_WMMA_F32_16X16X4_F32 per node.
        float xs[8];
#if HAS_WMMA_F32X4
        // A: 16x4, rows a = v2 vectors; lanes 0-15 hold (k0,k1), lanes 16-31 hold (k2,k3=0).
        v2f Af, Bf;
        Af.x = hi ? sV2[w][m][2] : sV2[w][m][0];
        Af.y = hi ? 0.f          : sV2[w][m][1];
        // B: 4x16, cols b = v1 vectors, mirrored layout (k3 row is zero-padded).
        Bf.x = hi ? sV1[w][m][2] : sV1[w][m][0];
        Bf.y = hi ? 0.f          : sV1[w][m][1];
        v8f Cf = {};
        v8f Df = __builtin_amdgcn_wmma_f32_16x16x4_f32(
            /*neg_a=*/false, Af, /*neg_b=*/false, Bf,
            /*c_mod=*/(short)0, Cf, /*reuse_a=*/false, /*reuse_b=*/false);
#pragma unroll
        for (int r = 0; r < 8; ++r) xs[r] = Df[r];
#else
#pragma unroll
        for (int r = 0; r < 8; ++r) {
            int a = r + (hi ? 8 : 0);
            xs[r] = sV2[w][a][0] * sV1[w][m][0] +
                    sV2[w][a][1] * sV1[w][m][1] +
                    sV2[w][a][2] * sV1[w][m][2];
        }
#endif
        // D layout: lane l, VGPR r -> row a = r + 8*(l>=16), col b = l&15.
#pragma unroll
        for (int r = 0; r < 8; ++r) {
            int a = r + (hi ? 8 : 0);
            int b = m;
            if (a < dout && b < din) {
                float x  = xs[r];
                float y2 = sN2[w][a] * sN1[w][b] - x * x;   // Lagrange identity
                float y  = sqrtf(fmaxf(y2, 0.f));
                float ang = atan2f(y, x);
                int rel = (int)(ang * (8.0f / PI_F));
                int t = st + a * din + b;
                out_edges[3 * t + 0] = sEin[w][a];
                out_edges[3 * t + 1] = sEout[w][b];
                out_edges[3 * t + 2] = rel;
                out_angle[t] = ang;
                out_w[t] = 1.0f;
            }
        }
    } else if (valid && din > 0 && dout > 0) {
        // Generic fallback for degree > 16: direct gathers + explicit cross product.
        int size = din * dout;
        for (int local = lane; local < size; local += 32) {
            int a = local / din;
            int b = local % din;
            int ein  = sorted_in[boo + a];
            int eout = sorted_out[boi + b];
            int ni = el[2 * eout + 1];
            int nk = el[2 * ein + 0];
            float v1x = pos[3 * ni + 0] - pjx, v1y = pos[3 * ni + 1] - pjy, v1z = pos[3 * ni + 2] - pjz;
            float v2x = pos[3 * nk + 0] - pjx, v2y = pos[3 * nk + 1] - pjy, v2z = pos[3 * nk + 2] - pjz;
            float x = v1x * v2x + v1y * v2y + v1z * v2z;
            float cx = v1y * v2z - v1z * v2y;
            float cy = v1z * v2x - v1x * v2z;
            float cz = v1x * v2y - v1y * v2x;
            float y = sqrtf(cx * cx + cy * cy + cz * cz);
            float ang = atan2f(y, x);
            int rel = (int)(ang * (8.0f / PI_F));
            int t = st + local;
            out_edges[3 * t + 0] = ein;
            out_edges[3 * t + 1] = eout;
            out_edges[3 * t + 2] = rel;
            out_angle[t] = ang;
            out_w[t] = 1.0f;
        }
    }
}

// ---------------- host launcher ----------------

extern "C" void kernel_launch(void* const* d_in, const int* in_sizes, int n_in,
                              void* d_out, int out_size, void* d_ws, size_t ws_size,
                              hipStream_t stream) {
    const int*   el  = (const int*)d_in[0];
    const float* pos = (const float*)d_in[1];
    const int E = in_sizes[0] / 2;
    const int N = in_sizes[1] / 3;
    const int L = out_size / 5;   // outputs: [L*3 int32 | L f32 angle | L f32 weight]
    (void)n_in; (void)ws_size; (void)L;

    // Workspace layout (int32):
    // [deg_in N][deg_out N][cnt_in N][cnt_out N][bo_in N][bo_out N][starts N][sorted_out E][sorted_in E]
    int* ws         = (int*)d_ws;
    int* deg_in     = ws + 0 * (size_t)N;
    int* deg_out    = ws + 1 * (size_t)N;
    int* cnt_in     = ws + 2 * (size_t)N;
    int* cnt_out    = ws + 3 * (size_t)N;
    int* bo_in      = ws + 4 * (size_t)N;
    int* bo_out     = ws + 5 * (size_t)N;
    int* starts     = ws + 6 * (size_t)N;
    int* sorted_out = ws + 7 * (size_t)N;
    int* sorted_in  = sorted_out + (size_t)E;

    int* out_edges   = (int*)d_out;
    float* out_angle = (float*)d_out + 3 * (size_t)L;
    float* out_w     = (float*)d_out + 4 * (size_t)L;

    const int B = 256;
    slg_zero_kernel<<<(4 * N + B - 1) / B, B, 0, stream>>>(deg_in, 4 * N);
    slg_count_kernel<<<(E + B - 1) / B, B, 0, stream>>>(el, E, deg_in, deg_out);
    slg_scan_kernel<<<1, B, 0, stream>>>(deg_in, deg_out, N, bo_in, bo_out, starts);
    slg_scatter_kernel<<<(E + B - 1) / B, B, 0, stream>>>(el, E, bo_in, bo_out,
                                                          cnt_in, cnt_out, sorted_out, sorted_in);
    slg_bucketsort_kernel<<<(N + B - 1) / B, B, 0, stream>>>(N, deg_in, deg_out, bo_in, bo_out,
                                                             sorted_out, sorted_in);
    slg_line_kernel<<<(N + WPB - 1) / WPB, B, 0, stream>>>(el, pos, N, deg_in, deg_out,
                                                           bo_in, bo_out, starts,
                                                           sorted_out, sorted_in,
                                                           out_edges, out_angle, out_w);
}